// Activs_prober_58317065945769
// MI455X (gfx1250) — compile-verified
//
#include <hip/hip_runtime.h>

// ---------------------------------------------------------------------------
// MI455X (gfx1250) fused kernel for:  (copy x) + per-row L2 norm ->
// sorted-segment mean per graph -> masked scalar mean.
//
// Roofline: 1 GB in + 1 GB out @ 23.3 TB/s ~= 86 us; compute ~0.5 GFLOP is
// negligible -> single-pass streaming kernel, NT cache hints, wave32 shuffle
// reductions. WMMA (V_WMMA_F32_16X16X4_F32) is used for the tiny 1024-bin
// final reduction via the ones-matrix trick (result invariant to A layout).
// ---------------------------------------------------------------------------

typedef float v4f __attribute__((ext_vector_type(4)));
typedef float v2f __attribute__((ext_vector_type(2)));
typedef float v8f __attribute__((ext_vector_type(8)));

#define FEAT_D          256          // feature dim (reference fixes D=256)
#define NBINS           1024         // NUM_GRAPHS
#define ROWS_PER_WAVE   16
#define WAVES_PER_BLOCK 8
#define ROWS_PER_BLOCK  (ROWS_PER_WAVE * WAVES_PER_BLOCK)   // 128 contiguous rows

// --------------------------------------------------------------------------
// Kernel 1: zero the 2*NBINS f32 accumulators (norm sums, counts) in d_ws.
// --------------------------------------------------------------------------
__global__ void zero_bins_kernel(float* __restrict__ bins) {
    int i = blockIdx.x * blockDim.x + threadIdx.x;
    if (i < 2 * NBINS) bins[i] = 0.0f;
}

// --------------------------------------------------------------------------
// Kernel 2: fused streaming pass.
//  - wave w of block b owns rows [ (b*8+w)*16 , +16 )  (contiguous => the
//    sorted batch ids within a wave form runs -> register run-aggregation,
//    ~2 global f32 atomics per wave instead of 32).
//  - per row: 32 lanes x 2 x float4 = 256 floats, NT load + NT store (copy),
//    sum of squares, 5-step wave32 xor-shuffle reduction, sqrt on lane 0.
// --------------------------------------------------------------------------
__global__ void fused_norm_copy_kernel(const float* __restrict__ x,
                                       const long long* __restrict__ batch,
                                       float* __restrict__ out,
                                       float* __restrict__ bins,   // [0..1023]=sum, [1024..2047]=count
                                       int nrows) {
    const int lane = threadIdx.x & 31;
    const int wave = threadIdx.x >> 5;                       // 0..7
    const long long rowBase =
        ((long long)blockIdx.x * WAVES_PER_BLOCK + wave) * ROWS_PER_WAVE;

    // lane-0 run aggregation state (batch is sorted)
    int   curBin = -1;
    float accSum = 0.0f;
    float accCnt = 0.0f;

    for (int i = 0; i < ROWS_PER_WAVE; ++i) {
        const long long row = rowBase + i;
        if (row >= nrows) break;                             // uniform per wave

        const v4f* __restrict__ src = (const v4f*)(x   + (size_t)row * FEAT_D);
        v4f*       __restrict__ dst = (v4f*)      (out + (size_t)row * FEAT_D);

        // coalesced: lanes cover cols [0,128) and [128,256) as float4s
        v4f a = __builtin_nontemporal_load(src + lane);
        v4f b = __builtin_nontemporal_load(src + lane + 32);
        __builtin_nontemporal_store(a, dst + lane);          // the "return input" copy
        __builtin_nontemporal_store(b, dst + lane + 32);

        float ss = a.x * a.x + a.y * a.y + a.z * a.z + a.w * a.w
                 + b.x * b.x + b.y * b.y + b.z * b.z + b.w * b.w;

        // wave32 all-reduce (warpSize == 32 on gfx1250)
        #pragma unroll
        for (int off = 16; off > 0; off >>= 1)
            ss += __shfl_xor(ss, off, 32);

        if (lane == 0) {
            const float nrm = sqrtf(ss);
            const int   bi  = (int)batch[row];
            if (bi != curBin) {
                if (curBin >= 0) {
                    atomicAdd(&bins[curBin],         accSum);
                    atomicAdd(&bins[NBINS + curBin], accCnt);
                }
                curBin = bi; accSum = nrm; accCnt = 1.0f;
            } else {
                accSum += nrm; accCnt += 1.0f;
            }
        }
    }
    if (lane == 0 && curBin >= 0) {
        atomicAdd(&bins[curBin],         accSum);
        atomicAdd(&bins[NBINS + curBin], accCnt);
    }
}

// --------------------------------------------------------------------------
// Kernel 3: single wave. per_graph[g] = cnt>0 ? sum/cnt : 0, masked by
// g < bmax (batch sorted -> bmax = batch[N-1]). The 1024-value reduction runs
// on the matrix pipe: D += A x Ones accumulates row sums of A; summing all 16
// D-rows (8 acc VGPRs + lane16 half via shuffle) yields sum of all A entries,
// independent of the exact A-operand lane permutation.
// --------------------------------------------------------------------------
__global__ void finalize_kernel(const float* __restrict__ bins,
                                const long long* __restrict__ batch,
                                float* __restrict__ out_scalar,
                                int nrows) {
    const int lane = threadIdx.x;               // blockDim.x == 32, EXEC all-1s
    const float bmax = (float)batch[nrows - 1]; // sorted ascending

    const int m  = lane & 15;
    const int kh = (lane >> 4) << 1;            // 0 or 2 (A 16x4 f32 layout)

    v2f ones; ones[0] = 1.0f; ones[1] = 1.0f;   // all-ones B (layout-invariant)
    v8f c = {};

    #pragma unroll
    for (int chunk = 0; chunk < NBINS / 64; ++chunk) {   // 16 WMMAs
        const int g0 = chunk * 64 + m * 4 + kh;
        v2f a;
        #pragma unroll
        for (int j = 0; j < 2; ++j) {
            const int g = g0 + j;
            const float s = bins[g];
            const float n = bins[NBINS + g];
            float pg = (n > 0.0f) ? (s / n) : 0.0f;      // where(counts>0, sum/cnt, 0)
            a[j] = ((float)g < bmax) ? pg : 0.0f;        // arange < bmax mask
        }
        // D = A x Ones + C   (v_wmma_f32_16x16x4_f32)
        c = __builtin_amdgcn_wmma_f32_16x16x4_f32(false, a, false, ones,
                                                  (short)0, c, false, false);
    }

    // lanes 0-15 hold rowsums of M=0..7 in c[0..7]; lanes 16-31 hold M=8..15
    float s = c[0] + c[1] + c[2] + c[3] + c[4] + c[5] + c[6] + c[7];
    s += __shfl_xor(s, 16, 32);                 // add the other half's rows

    if (lane == 0) *out_scalar = s / (bmax + 1.0f);
}

// --------------------------------------------------------------------------
extern "C" void kernel_launch(void* const* d_in, const int* in_sizes, int n_in,
                              void* d_out, int out_size, void* d_ws, size_t ws_size,
                              hipStream_t stream) {
    const float*     x     = (const float*)d_in[0];       // [N, 256] f32
    const long long* batch = (const long long*)d_in[1];   // [N] int64, sorted
    float*           out   = (float*)d_out;               // [N*256 copy][1 scalar]
    float*           bins  = (float*)d_ws;                // 2*1024 f32 accumulators

    const int N = in_sizes[1];

    // 1) clear accumulators (fresh every call; stream-ordered)
    zero_bins_kernel<<<(2 * NBINS + 255) / 256, 256, 0, stream>>>(bins);

    // 2) fused copy + row-norm + sorted-segment accumulation
    const int blocks = (N + ROWS_PER_BLOCK - 1) / ROWS_PER_BLOCK;
    fused_norm_copy_kernel<<<blocks, WAVES_PER_BLOCK * 32, 0, stream>>>(
        x, batch, out, bins, N);

    // 3) WMMA-based final reduction -> scalar at out[N*D]
    finalize_kernel<<<1, 32, 0, stream>>>(bins, batch,
                                          out + (size_t)N * FEAT_D, N);
}